// Renderer_2851858284922
// MI455X (gfx1250) — compile-verified
//
#include <hip/hip_runtime.h>
#include <math.h>

typedef __attribute__((ext_vector_type(2))) float v2f;
typedef __attribute__((ext_vector_type(8))) float v8f;

#define IMG 256
#define NVERT 4096
#define NFACE 1024
#define NEARZ 0.1f
#define FARZ 100.0f
#define INV_SIGMA 100.0f
#define LOG2E 1.44269504088896340736f
#define TANW 0.57735026918962576451f  // tan(30 deg)

// ---------------------------------------------------------------------------
// Kernel 1: per-face preprocessing.
// Camera chain collapses to (x, y, -z) then perspective divide by (z * tan30).
// Emits, per face, the 3 edge-line equations as affine coefficients in
// (PX, PY, 1), plus base-2-prescaled scale/bias for the sigmoid:
//   scale = valid ? sign(area2)/sigma * log2(e) : 0
//   bias  = valid ? 0 : -90   (exp2(-90) ~ 0 -> factor 1 for culled faces)
// Workspace layout (float, stride S = B*NFACE):
//   slot e*3+k  (e=edge 0..2, k=coef 0..2)  -> 9 slots
//   slot 9 = scale, slot 10 = bias
// ---------------------------------------------------------------------------
__global__ void prep_kernel(const float* __restrict__ verts,
                            const int* __restrict__ faces,
                            float* __restrict__ ws, int B) {
  int gid = blockIdx.x * blockDim.x + threadIdx.x;
  int total = B * NFACE;
  if (gid >= total) return;
  int b = gid / NFACE;
  const int* fi = faces + gid * 3;

  float fx[3], fy[3], fz[3];
#pragma unroll
  for (int i = 0; i < 3; i++) {
    int vi = fi[i];
    const float* v = verts + (b * NVERT + vi) * 3;
    float X = v[0], Y = v[1], Z = -v[2];
    float invzw = 1.0f / (Z * TANW);
    fx[i] = X * invzw;
    fy[i] = Y * invzw;
    fz[i] = Z;
  }

  float area2 = (fx[1] - fx[0]) * (fy[2] - fy[0]) -
                (fy[1] - fy[0]) * (fx[2] - fx[0]);
  float s = (area2 > 0.0f) ? 1.0f : ((area2 < 0.0f) ? -1.0f : 0.0f);
  float zmin = fminf(fz[0], fminf(fz[1], fz[2]));
  float zmax = fmaxf(fz[0], fmaxf(fz[1], fz[2]));
  bool valid = (zmin > NEARZ) && (zmax < FARZ);

  const int S = total;
  const int ei[3] = {0, 1, 2};
  const int ej[3] = {1, 2, 0};
#pragma unroll
  for (int e = 0; e < 3; e++) {
    int i = ei[e], j = ej[e];
    float ex = fx[j] - fx[i];
    float ey = fy[j] - fy[i];
    float L = sqrtf(ex * ex + ey * ey) + 1e-12f;
    float invL = 1.0f / L;
    // d = ex*(PY - y0) - ey*(PX - x0) = c0*PX + c1*PY + c2  (then /L)
    ws[(e * 3 + 0) * S + gid] = -ey * invL;
    ws[(e * 3 + 1) * S + gid] = ex * invL;
    ws[(e * 3 + 2) * S + gid] = (ey * fx[i] - ex * fy[i]) * invL;
  }
  ws[9 * S + gid] = valid ? s * INV_SIGMA * LOG2E : 0.0f;
  ws[10 * S + gid] = valid ? 0.0f : -90.0f;
}

// ---------------------------------------------------------------------------
// Kernel 2: rasterize via V_WMMA_F32_16X16X4_F32.
// One wave = 64 pixels of one image row (4 M-tiles of 16 pixels).
// A (16x4, per M-tile): lanes 0-15 hold (PX_m, PY) for K=0,1; lanes 16-31
// hold (1, 0) for K=2,3 (per ISA 32-bit A layout).
// B (4x16, per edge): lanes 0-15 hold (c0, c1) of face col N=lane; lanes
// 16-31 hold (c2, 0).
// C/D layout: lane owns face column N = lane&15; VGPR r = pixel row r (+8
// for upper half) -> per-face scale/bias is one scalar per lane.
//
// Accumulation: Pi 1/(1+2^t) = 1 / Pi (1+2^t); accumulate the growing
// product (factors >= 1; overflow to +inf is the correct limit) and take a
// single reciprocal at the end. acc*(1+e) is fused as fma(e, acc, acc), so
// the per-cell cost is: min3 + 1/2 pk_fma + fma (VALU) and one v_exp_f32
// (TRANS). Product over the 16 face columns via __shfl_xor (masks 8..1 stay
// inside each 16-lane half).
// ---------------------------------------------------------------------------
__global__ void raster_kernel(const float* __restrict__ ws,
                              float* __restrict__ out, int B) {
  const int S = B * NFACE;
  int wave = (blockIdx.x * blockDim.x + threadIdx.x) >> 5;
  int lane = threadIdx.x & 31;

  int strip = wave & 3;              // 4 strips of 64 pixels per row
  int y = (wave >> 2) & (IMG - 1);   // image row
  int b = wave >> 10;                // batch (IMG/64 * IMG = 1024 waves/batch)
  int x0 = strip * 64;

  bool hi = lane >= 16;
  int lcol = lane & 15;

  float PY = -((2.0f * y + 1.0f - (float)IMG) / (float)IMG);

  v2f a[4];
#pragma unroll
  for (int t = 0; t < 4; t++) {
    float px = (2.0f * (x0 + t * 16 + lcol) + 1.0f - (float)IMG) / (float)IMG;
    a[t].x = hi ? 1.0f : px;   // K=2 coeff (the "1" column) on upper half
    a[t].y = hi ? 0.0f : PY;   // K=3 is zero padding
  }

  v8f acc[4];
#pragma unroll
  for (int t = 0; t < 4; t++)
#pragma unroll
    for (int r = 0; r < 8; r++) acc[t][r] = 1.0f;

  for (int fc = 0; fc < NFACE; fc += 16) {
    int idx = b * NFACE + fc + lcol;

    v2f bmat[3];
#pragma unroll
    for (int e = 0; e < 3; e++) {
      float c0 = ws[(e * 3 + 0) * S + idx];
      float c1 = ws[(e * 3 + 1) * S + idx];
      float c2 = ws[(e * 3 + 2) * S + idx];
      bmat[e].x = hi ? c2 : c0;
      bmat[e].y = hi ? 0.0f : c1;
    }
    float sc = ws[9 * S + idx];
    float bi = ws[10 * S + idx];

    v8f czero = {0.0f, 0.0f, 0.0f, 0.0f, 0.0f, 0.0f, 0.0f, 0.0f};
#pragma unroll
    for (int t = 0; t < 4; t++) {
      v8f d0 = __builtin_amdgcn_wmma_f32_16x16x4_f32(
          false, a[t], false, bmat[0], (short)0, czero, false, false);
      v8f d1 = __builtin_amdgcn_wmma_f32_16x16x4_f32(
          false, a[t], false, bmat[1], (short)0, czero, false, false);
      v8f d2 = __builtin_amdgcn_wmma_f32_16x16x4_f32(
          false, a[t], false, bmat[2], (short)0, czero, false, false);
#pragma unroll
      for (int r = 0; r < 8; r++) {
        float m = fminf(fminf(d0[r], d1[r]), d2[r]);
        float tt = fmaf(m, sc, bi);                 // log2 domain
        float e2 = __builtin_amdgcn_exp2f(tt);      // v_exp_f32
        // acc *= (1 + e2)  ==  fma(e2, acc, acc)   (single VALU op)
        acc[t][r] = fmaf(e2, acc[t][r], acc[t][r]);
      }
    }
  }

  // product-reduce over the 16 face columns inside each 16-lane half
#pragma unroll
  for (int mask = 8; mask >= 1; mask >>= 1) {
#pragma unroll
    for (int t = 0; t < 4; t++)
#pragma unroll
      for (int r = 0; r < 8; r++)
        acc[t][r] *= __shfl_xor(acc[t][r], mask, 32);
  }

  int obase = b * IMG * IMG + y * IMG + x0;
#pragma unroll
  for (int t = 0; t < 4; t++)
#pragma unroll
    for (int r = 0; r < 8; r++)
      if (lcol == r)
        out[obase + t * 16 + r + (hi ? 8 : 0)] =
            1.0f - __builtin_amdgcn_rcpf(acc[t][r]);
}

// ---------------------------------------------------------------------------
extern "C" void kernel_launch(void* const* d_in, const int* in_sizes, int n_in,
                              void* d_out, int out_size, void* d_ws,
                              size_t ws_size, hipStream_t stream) {
  const float* verts = (const float*)d_in[0];
  const int* faces = (const int*)d_in[1];
  float* out = (float*)d_out;
  float* ws = (float*)d_ws;

  int B = in_sizes[0] / (NVERT * 3);   // B = 2 for the reference setup

  int total_faces = B * NFACE;
  prep_kernel<<<(total_faces + 255) / 256, 256, 0, stream>>>(verts, faces, ws,
                                                             B);

  int waves = B * IMG * (IMG / 64);    // one wave per 64-pixel strip
  int threads = waves * 32;
  raster_kernel<<<threads / 256, 256, 0, stream>>>(ws, out, B);
}